// Decouple_18442589569620
// MI455X (gfx1250) — compile-verified
//
#include <hip/hip_runtime.h>

typedef __attribute__((ext_vector_type(2))) float v2f;
typedef __attribute__((ext_vector_type(8))) float v8f;

#define BATCH 2
#define CH    64
#define IH    192
#define IW    192
#define HWSZ  (IH * IW)          // 36864
#define NPIX  (BATCH * HWSZ)     // 73728
#define EXPF  4
#define CE    (CH * EXPF)        // 256
#define DGRP  4
#define KK9   9
#define NOFF  (DGRP * 2 * KK9)   // 72
#define XDPAD (CE + 4)           // 260: A-frag ds_load_b64 conflict-free (4*m banks)
#define TSTR  84                 // t transpose tile stride: 8*84 % 64 = 32 (halves disjoint)
#define MSTR  69                 // m transpose tile stride: coprime-ish, conflict-free R/W

// ---------------------------------------------------------------------------
// Kernel 2: fused diff-depthwise-3x3 expansion + 1x1 (256->72) GEMM via WMMA.
// 64 threads = 2 waves per workgroup; 32 pixels (one row segment) per block.
// ---------------------------------------------------------------------------
template <bool INTERIOR>
__device__ __forceinline__ void stage_xd(const float* __restrict__ x,
                                         const float* __restrict__ lds_w1,
                                         float* __restrict__ lds_xd,
                                         int b0, int yy, int xxp, int rem,
                                         int pl, int w0)
{
    for (int i = 0; i < 32; ++i) {
        const int c = w0 + 2 * i;                     // wave-uniform channel
        const float* __restrict__ xp = x + (size_t)(b0 * CH + c) * HWSZ;
        const float* __restrict__ lw = lds_w1 + c * 36;   // broadcast reads
        const float xc = xp[rem];
        float a0 = 0.f, a1 = 0.f, a2 = 0.f, a3 = 0.f;
#pragma unroll
        for (int tap = 0; tap < 9; ++tap) {
            if (tap == 4) continue;                   // center tap: diff == 0
            const int dy = tap / 3 - 1, dx = tap % 3 - 1;
            float nv;
            if (INTERIOR) {
                nv = xp[rem + dy * IW + dx];          // unconditional, const offset
            } else {
                const int ny = yy + dy, nx = xxp + dx;
                const bool valid = ((unsigned)ny < IH) && ((unsigned)nx < IW);
                const int nyc = ny < 0 ? 0 : (ny > IH - 1 ? IH - 1 : ny);
                const int nxc = nx < 0 ? 0 : (nx > IW - 1 ? IW - 1 : nx);
                const float l = xp[nyc * IW + nxc];   // clamped, unconditional
                nv = valid ? l : 0.f;                 // v_cndmask, no branch
            }
            const float d = nv - xc;
            a0 += lw[0 * 9 + tap] * d;
            a1 += lw[1 * 9 + tap] * d;
            a2 += lw[2 * 9 + tap] * d;
            a3 += lw[3 * 9 + tap] * d;
        }
        float* dst = lds_xd + pl * XDPAD + c * 4;     // 16B-aligned ds_store_b128
        dst[0] = a0; dst[1] = a1; dst[2] = a2; dst[3] = a3;
    }
}

__global__ __launch_bounds__(64)
void offset_gemm_kernel(const float* __restrict__ x,
                        const float* __restrict__ w1,   // (256, 3, 3)
                        const float* __restrict__ w2,   // (72, 256)
                        const float* __restrict__ b2,   // (72)
                        float* __restrict__ t_ws)       // (NPIX, 72) pixel-major
{
    __shared__ float lds_xd[32 * XDPAD];
    __shared__ float lds_w1[CE * KK9];                  // 2304 floats

    const int tid   = threadIdx.x;
    const int pbase = blockIdx.x * 32;

    __builtin_prefetch(w2 + tid * 288, 0, 1);           // global_prefetch_b8

    for (int j = tid; j < CE * KK9; j += 64) lds_w1[j] = w1[j];
    __syncthreads();

    // geometry (block-uniform): 32-pixel tile inside one image row
    const int b0   = pbase / HWSZ;
    const int rem0 = pbase - b0 * HWSZ;
    const int yy   = rem0 / IW;
    const int xx0  = rem0 - yy * IW;
    const bool interior =
        (yy > 0) && (yy < IH - 1) && (xx0 != 0) && (xx0 + 32 != IW);

    const int pl  = tid & 31;                           // lane <-> pixel: coalesced
    const int w0  = tid >> 5;
    const int rem = rem0 + pl;
    const int xxp = xx0 + pl;

    if (interior) stage_xd<true >(x, lds_w1, lds_xd, b0, yy, xxp, rem, pl, w0);
    else          stage_xd<false>(x, lds_w1, lds_xd, b0, yy, xxp, rem, pl, w0);
    __syncthreads();

    // ---- GEMM: A = xd (16 pix x 256), B = w2^T (256 x 80), f32 WMMA ----
    const int  lane  = tid & 31;
    const bool hi    = lane >= 16;
    const int  lm    = lane & 15;
    const int  mbase = w0 * 16;
    const int  arow  = (mbase + lm) * XDPAD;

    int bof[5];
#pragma unroll
    for (int nt = 0; nt < 5; ++nt) {
        int o = nt * 16 + lm;
        if (o > NOFF - 1) o = NOFF - 1;                 // clamp: cols >=72 unused
        bof[nt] = o * CE + (hi ? 2 : 0);
    }

    v8f acc[5];
    v8f zero = {};
#pragma unroll
    for (int nt = 0; nt < 5; ++nt) acc[nt] = zero;

    for (int kb = 0; kb < CE; kb += 4) {
        const int ka = kb + (hi ? 2 : 0);
        const v2f afrag = *(const v2f*)(lds_xd + arow + ka);   // ds_load_b64
#pragma unroll
        for (int nt = 0; nt < 5; ++nt) {
            const v2f bfrag = *(const v2f*)(w2 + bof[nt] + kb); // global_load_b64
            acc[nt] = __builtin_amdgcn_wmma_f32_16x16x4_f32(
                false, afrag, false, bfrag, (short)0, acc[nt], false, false);
        }
    }

    // ---- transpose via LDS (reuse xd tile), then coalesced global stores ----
    __syncthreads();
    float* lds_t = lds_xd;                              // [32][TSTR]
#pragma unroll
    for (int nt = 0; nt < 5; ++nt) {
        const int o = nt * 16 + lm;
        if (o < NOFF) {
#pragma unroll
            for (int v = 0; v < 8; ++v) {
                const int m = v + (hi ? 8 : 0);
                lds_t[(mbase + m) * TSTR + o] = acc[nt][v];
            }
        }
    }
    __syncthreads();
    for (int idx = tid; idx < 32 * NOFF; idx += 64) {
        const int p = idx / NOFF;
        const int o = idx - p * NOFF;
        t_ws[(size_t)(pbase + p) * NOFF + o] = lds_t[p * TSTR + o] + b2[o];
    }
}

// ---------------------------------------------------------------------------
// Kernel 3: depthwise deformable 3x3 (bilinear) + center correction.
// One block per pixel, one thread per channel. All gathers unconditional
// (clamped address + cndmask), matching the reference clamp+mask semantics.
// ---------------------------------------------------------------------------
__global__ __launch_bounds__(64)
void deform_kernel(const float* __restrict__ x,
                   const float* __restrict__ t_ws,  // (NPIX, 72)
                   const float* __restrict__ wd,    // (64, 3, 3)
                   float* __restrict__ y_ws)        // (NPIX, 64) pixel-major
{
    __shared__ float toff[NOFF];
    const int p = blockIdx.x;
    const int c = threadIdx.x;

    toff[c] = t_ws[(size_t)p * NOFF + c];
    if (c < NOFF - 64) toff[64 + c] = t_ws[(size_t)p * NOFF + 64 + c];
    __syncthreads();

    const int b   = p / HWSZ;
    const int rem = p - b * HWSZ;
    const int yy  = rem / IW;
    const int xx  = rem - yy * IW;
    const int g   = c >> 4;                          // deform group (Cg = 16)
    const float* __restrict__ xp = x + (size_t)(b * CH + c) * HWSZ;

    float acc = 0.f, wsum = 0.f;
#pragma unroll
    for (int kk = 0; kk < KK9; ++kk) {
        const float wv = wd[c * KK9 + kk];
        wsum += wv;
        const float dy = toff[g * 18 + kk * 2 + 0];
        const float dx = toff[g * 18 + kk * 2 + 1];
        const float py = (float)(yy - 1 + kk / 3) + dy;
        const float px = (float)(xx - 1 + kk % 3) + dx;
        const float fy = floorf(py), fx = floorf(px);
        const int   iy = (int)fy,    ix = (int)fx;
        const float wy = py - fy,    wx = px - fx;

        const bool vy0 = (unsigned)iy       < IH;
        const bool vy1 = (unsigned)(iy + 1) < IH;
        const bool vx0 = (unsigned)ix       < IW;
        const bool vx1 = (unsigned)(ix + 1) < IW;
        const int iy0 = iy     < 0 ? 0 : (iy     > IH - 1 ? IH - 1 : iy);
        const int iy1 = iy + 1 < 0 ? 0 : (iy + 1 > IH - 1 ? IH - 1 : iy + 1);
        const int ix0 = ix     < 0 ? 0 : (ix     > IW - 1 ? IW - 1 : ix);
        const int ix1 = ix + 1 < 0 ? 0 : (ix + 1 > IW - 1 ? IW - 1 : ix + 1);

        const float l00 = xp[iy0 * IW + ix0];
        const float l01 = xp[iy0 * IW + ix1];
        const float l10 = xp[iy1 * IW + ix0];
        const float l11 = xp[iy1 * IW + ix1];
        const float v00 = (vy0 && vx0) ? l00 : 0.f;
        const float v01 = (vy0 && vx1) ? l01 : 0.f;
        const float v10 = (vy1 && vx0) ? l10 : 0.f;
        const float v11 = (vy1 && vx1) ? l11 : 0.f;

        const float val = v00 * (1.f - wy) * (1.f - wx)
                        + v01 * (1.f - wy) * wx
                        + v10 * wy * (1.f - wx)
                        + v11 * wy * wx;
        acc += val * wv;
    }
    acc -= xp[rem] * wsum;                           // y -= x * sum(w_def)
    y_ws[(size_t)p * CH + c] = acc;
}

// ---------------------------------------------------------------------------
// Kernel 4: final 1x1 conv 64->64 as WMMA f32 GEMM (73728 x 64 x 64),
// LDS-transposed so NCHW stores are coalesced.
// ---------------------------------------------------------------------------
__global__ __launch_bounds__(64)
void out_gemm_kernel(const float* __restrict__ y_ws,  // (NPIX, 64) row-major A
                     const float* __restrict__ wout,  // (64, 64) [o][c]
                     float* __restrict__ out_m)       // (B, 64, H, W)
{
    __shared__ float lds_m[32 * MSTR];

    const int  tid   = threadIdx.x;
    const int  wave  = tid >> 5;
    const int  lane  = tid & 31;
    const bool hi    = lane >= 16;
    const int  lm    = lane & 15;
    const int  pb    = blockIdx.x * 32;
    const int  pbase = pb + wave * 16;

    __builtin_prefetch(wout + tid * 64, 0, 1);        // global_prefetch_b8

    int bof[4];
#pragma unroll
    for (int nt = 0; nt < 4; ++nt)
        bof[nt] = (nt * 16 + lm) * CH + (hi ? 2 : 0);

    v8f acc[4];
    v8f zero = {};
#pragma unroll
    for (int nt = 0; nt < 4; ++nt) acc[nt] = zero;

    for (int kb = 0; kb < CH; kb += 4) {
        const int ka = kb + (hi ? 2 : 0);
        const v2f afrag = *(const v2f*)(y_ws + (size_t)(pbase + lm) * CH + ka);
#pragma unroll
        for (int nt = 0; nt < 4; ++nt) {
            const v2f bfrag = *(const v2f*)(wout + bof[nt] + kb);
            acc[nt] = __builtin_amdgcn_wmma_f32_16x16x4_f32(
                false, afrag, false, bfrag, (short)0, acc[nt], false, false);
        }
    }

    // transpose through LDS, then coalesced NCHW stores
#pragma unroll
    for (int nt = 0; nt < 4; ++nt) {
        const int o = nt * 16 + lm;
#pragma unroll
        for (int v = 0; v < 8; ++v) {
            const int m = v + (hi ? 8 : 0);
            lds_m[(wave * 16 + m) * MSTR + o] = acc[nt][v];
        }
    }
    __syncthreads();

    const int b0   = pb / HWSZ;
    const int rem0 = pb - b0 * HWSZ;
    for (int idx = tid; idx < 32 * CH; idx += 64) {
        const int o  = idx >> 5;
        const int p  = idx & 31;
        out_m[(size_t)(b0 * CH + o) * HWSZ + rem0 + p] = lds_m[p * MSTR + o];
    }
}

// ---------------------------------------------------------------------------
extern "C" void kernel_launch(void* const* d_in, const int* in_sizes, int n_in,
                              void* d_out, int out_size, void* d_ws, size_t ws_size,
                              hipStream_t stream)
{
    const float* x1     = (const float*)d_in[0];
    const float* w_off1 = (const float*)d_in[1];
    const float* w_off2 = (const float*)d_in[2];
    const float* b_off2 = (const float*)d_in[3];
    const float* w_def  = (const float*)d_in[4];
    const float* w_out  = (const float*)d_in[5];

    float* out  = (float*)d_out;
    float* ws   = (float*)d_ws;
    float* t_ws = ws;                               // NPIX * 72 floats
    float* y_ws = ws + (size_t)NPIX * NOFF;         // NPIX * 64 floats

    // output[0]: x1 passthrough (graph-capture-safe async D2D copy)
    hipMemcpyAsync(out, x1, sizeof(float) * (size_t)NPIX * CH,
                   hipMemcpyDeviceToDevice, stream);
    float* out_m = out + (size_t)NPIX * CH;

    offset_gemm_kernel<<<NPIX / 32, 64, 0, stream>>>(x1, w_off1, w_off2,
                                                     b_off2, t_ws);
    deform_kernel<<<NPIX, 64, 0, stream>>>(x1, t_ws, w_def, y_ws);
    out_gemm_kernel<<<NPIX / 32, 64, 0, stream>>>(y_ws, w_out, out_m);
}